// GNN_layers_63745904607323
// MI455X (gfx1250) — compile-verified
//
#include <hip/hip_runtime.h>

#define C_DIM 64
#define G_DIM 256

typedef __attribute__((ext_vector_type(2))) float v2f;
typedef __attribute__((ext_vector_type(8))) float v8f;

// ---------------- zero-fill (stats/pool accumulators live in poisoned ws) ---
__global__ void k_fill0(float* __restrict__ p, int n) {
  int i = blockIdx.x * blockDim.x + threadIdx.x;
  if (i < n) p[i] = 0.0f;
}

// ---------------- agg = (1+eps)*x = 2*x ------------------------------------
__global__ void k_init2x(const float4* __restrict__ x4, float4* __restrict__ a4, int n4) {
  int i = blockIdx.x * blockDim.x + threadIdx.x;
  if (i >= n4) return;
  float4 v = x4[i];
  float4 o;
  o.x = 2.0f * v.x; o.y = 2.0f * v.y; o.z = 2.0f * v.z; o.w = 2.0f * v.w;
  a4[i] = o;
}

// ---------------- edge scatter-add: agg[dst] += x[src], 16 lanes per edge ---
__global__ void k_scatter(const float4* __restrict__ x4, const int* __restrict__ src,
                          const int* __restrict__ dst, float* __restrict__ agg, int E_) {
  int t = blockIdx.x * blockDim.x + threadIdx.x;
  int e = t >> 4;
  if (e >= E_) return;
  int q = t & 15;
  float4 v = x4[(size_t)src[e] * 16 + q];
  float* o = agg + (size_t)dst[e] * C_DIM + (size_t)q * 4;
  atomicAdd(o + 0, v.x);
  atomicAdd(o + 1, v.y);
  atomicAdd(o + 2, v.z);
  atomicAdd(o + 3, v.w);
}

// ---------------- out[N,64] = in[N,64] @ W[64,64] + bias, via WMMA f32 ------
// One wave per 16-row tile. K=64 -> 16 steps of V_WMMA_F32_16X16X4_F32 over
// 4 N-tiles (16 cols each) = 64 WMMA / wave.
// W is staged TRANSPOSED in LDS (sWt[col*68 + k]) so each B fragment
// {W[k0][col], W[k0+1][col]} is one contiguous, 8B-aligned ds_load_b64 into
// an even VGPR pair -- no packing movs in the WMMA loop. Row stride 68 keeps
// all fragment reads bank-conflict-free (bank = (4*lane + k0) & 63).
__launch_bounds__(256)
__global__ void k_gemm64(const float* __restrict__ in, const float* __restrict__ W,
                         const float* __restrict__ bias, float* __restrict__ out,
                         int nTiles) {
  __shared__ float sWt[64 * 68];      // [col][k], stride 68
  __shared__ float sA[8][16 * 68];    // per-wave [row][k], stride 68
  const int tid  = threadIdx.x;
  const int wave = tid >> 5;
  const int lane = tid & 31;

  // stage W (64x64 row-major in memory) -> transposed LDS
  {
    const float4* Wf4 = (const float4*)W;
#pragma unroll
    for (int i = 0; i < 4; ++i) {
      int q = tid + i * 256;            // float4 index 0..1023
      int row = q >> 4;                 // k
      int cb  = (q & 15) * 4;           // col base
      float4 v = Wf4[q];
      sWt[(cb + 0) * 68 + row] = v.x;
      sWt[(cb + 1) * 68 + row] = v.y;
      sWt[(cb + 2) * 68 + row] = v.z;
      sWt[(cb + 3) * 68 + row] = v.w;
    }
  }

  const int tile    = blockIdx.x * 8 + wave;
  const bool active = tile < nTiles;   // uniform per wave: EXEC stays all-1s
  if (active) {
    const int r  = lane >> 1;          // 16 rows, 2 lanes per row
    const int cb = (lane & 1) * 32;
    const float4* srcp = (const float4*)(in + (size_t)(tile * 16 + r) * C_DIM + cb);
    float4* dstp = (float4*)&sA[wave][r * 68 + cb];
#pragma unroll
    for (int i = 0; i < 8; ++i) dstp[i] = srcp[i];
  }
  __syncthreads();
  if (!active) return;

  const int half = lane >> 4;          // ISA 16x4 A layout: lanes>=16 hold K+2
  const int l    = lane & 15;

  v8f acc[4] = {};
  const float* As = &sA[wave][l * 68];
  const float* Bt0 = &sWt[(l + 0)  * 68];
  const float* Bt1 = &sWt[(l + 16) * 68];
  const float* Bt2 = &sWt[(l + 32) * 68];
  const float* Bt3 = &sWt[(l + 48) * 68];
#pragma unroll
  for (int kk = 0; kk < 16; ++kk) {
    const int k0 = kk * 4 + half * 2;  // even -> 8B-aligned LDS v2f loads
    v2f a  = *(const v2f*)(As  + k0);
    v2f b0 = *(const v2f*)(Bt0 + k0);
    v2f b1 = *(const v2f*)(Bt1 + k0);
    v2f b2 = *(const v2f*)(Bt2 + k0);
    v2f b3 = *(const v2f*)(Bt3 + k0);
    acc[0] = __builtin_amdgcn_wmma_f32_16x16x4_f32(false, a, false, b0, (short)0, acc[0], false, false);
    acc[1] = __builtin_amdgcn_wmma_f32_16x16x4_f32(false, a, false, b1, (short)0, acc[1], false, false);
    acc[2] = __builtin_amdgcn_wmma_f32_16x16x4_f32(false, a, false, b2, (short)0, acc[2], false, false);
    acc[3] = __builtin_amdgcn_wmma_f32_16x16x4_f32(false, a, false, b3, (short)0, acc[3], false, false);
  }

  // D layout: vgpr j, lanes<16 -> row j, lanes>=16 -> row j+8; col = lane&15
  float* op = out + (size_t)(tile * 16 + half * 8) * C_DIM + l;
#pragma unroll
  for (int nt = 0; nt < 4; ++nt) {
    const float bl = bias[nt * 16 + l];
#pragma unroll
    for (int j = 0; j < 8; ++j)
      op[(size_t)j * C_DIM + nt * 16] = acc[nt][j] + bl;
  }
}

// ---------------- per-graph LN pass 1: sum, sum^2 (and counts) --------------
__global__ void k_ln_stats(const float4* __restrict__ h4, const int* __restrict__ batch,
                           float* __restrict__ sum, float* __restrict__ sumsq,
                           float* __restrict__ counts, int N_) {
  int n = blockIdx.x * blockDim.x + threadIdx.x;
  if (n >= N_) return;
  const float4* r = h4 + (size_t)n * 16;
  float s = 0.0f, q = 0.0f;
#pragma unroll
  for (int i = 0; i < 16; ++i) {
    float4 v = r[i];
    s += v.x + v.y + v.z + v.w;
    q += v.x * v.x + v.y * v.y + v.z * v.z + v.w * v.w;
  }
  int g = batch[n];
  atomicAdd(&sum[g], s);
  atomicAdd(&sumsq[g], q);
  if (counts) atomicAdd(&counts[g], 1.0f);
}

// ---------------- LN finalize: mean, 1/sqrt(var+eps) per graph --------------
__global__ void k_ln_finalize(const float* __restrict__ sum, const float* __restrict__ sumsq,
                              const float* __restrict__ counts,
                              float* __restrict__ mean, float* __restrict__ rsig) {
  int g = blockIdx.x * blockDim.x + threadIdx.x;
  if (g >= G_DIM) return;
  float norm = fmaxf(counts[g], 1.0f) * (float)C_DIM;
  float m = sum[g] / norm;
  float v = sumsq[g] / norm - m * m;
  mean[g] = m;
  rsig[g] = rsqrtf(fmaxf(v, 0.0f) + 1e-5f);
}

// ---------------- x1 = x + relu(LN(h)) --------------------------------------
__global__ void k_apply_ln1(const float4* __restrict__ x4, const float4* __restrict__ h4,
                            const int* __restrict__ batch,
                            const float* __restrict__ mean, const float* __restrict__ rsig,
                            const float* __restrict__ w, const float* __restrict__ b,
                            float4* __restrict__ out4, int n4) {
  int i = blockIdx.x * blockDim.x + threadIdx.x;
  if (i >= n4) return;
  int g = batch[i >> 4];
  float m = mean[g], rs = rsig[g], W_ = w[0], B_ = b[0];
  float4 h = h4[i], x = x4[i], o;
  o.x = x.x + fmaxf((h.x - m) * rs * W_ + B_, 0.0f);
  o.y = x.y + fmaxf((h.y - m) * rs * W_ + B_, 0.0f);
  o.z = x.z + fmaxf((h.z - m) * rs * W_ + B_, 0.0f);
  o.w = x.w + fmaxf((h.w - m) * rs * W_ + B_, 0.0f);
  out4[i] = o;
}

// ---------------- x_out = relu(LN(h2)); pool[g] += x_out --------------------
__global__ void k_apply_ln2_pool(const float4* __restrict__ h4, const int* __restrict__ batch,
                                 const float* __restrict__ mean, const float* __restrict__ rsig,
                                 const float* __restrict__ w, const float* __restrict__ b,
                                 float4* __restrict__ out4, float* __restrict__ pool, int n4) {
  int i = blockIdx.x * blockDim.x + threadIdx.x;
  if (i >= n4) return;
  int g = batch[i >> 4];
  float m = mean[g], rs = rsig[g], W_ = w[0], B_ = b[0];
  float4 h = h4[i], o;
  o.x = fmaxf((h.x - m) * rs * W_ + B_, 0.0f);
  o.y = fmaxf((h.y - m) * rs * W_ + B_, 0.0f);
  o.z = fmaxf((h.z - m) * rs * W_ + B_, 0.0f);
  o.w = fmaxf((h.w - m) * rs * W_ + B_, 0.0f);
  out4[i] = o;
  float* p = pool + (size_t)g * C_DIM + (size_t)(i & 15) * 4;
  atomicAdd(p + 0, o.x);
  atomicAdd(p + 1, o.y);
  atomicAdd(p + 2, o.z);
  atomicAdd(p + 3, o.w);
}

// ---------------- graph_emds = pool / counts --------------------------------
__global__ void k_pool_div(const float* __restrict__ pool, const float* __restrict__ counts,
                           float* __restrict__ out, int GC) {
  int i = blockIdx.x * blockDim.x + threadIdx.x;
  if (i >= GC) return;
  out[i] = pool[i] / fmaxf(counts[i >> 6], 1.0f);
}

extern "C" void kernel_launch(void* const* d_in, const int* in_sizes, int n_in,
                              void* d_out, int out_size, void* d_ws, size_t ws_size,
                              hipStream_t stream) {
  (void)n_in; (void)out_size; (void)ws_size;

  const float* x    = (const float*)d_in[0];
  const int*   ei   = (const int*)d_in[1];     // [2,E] flat: src then dst
  const int*   batch= (const int*)d_in[2];
  const float* Wg   = (const float*)d_in[3];
  const float* bg   = (const float*)d_in[4];
  const float* l1w  = (const float*)d_in[5];
  const float* l1b  = (const float*)d_in[6];
  const float* Wf   = (const float*)d_in[7];
  const float* bf   = (const float*)d_in[8];
  const float* l2w  = (const float*)d_in[9];
  const float* l2b  = (const float*)d_in[10];
  float* out = (float*)d_out;

  const int N_ = in_sizes[0] / C_DIM;          // 100000
  const int E_ = in_sizes[1] / 2;              // 1200000
  const int n4 = N_ * (C_DIM / 4);             // float4 elements in [N,C]
  const int nTiles = (N_ + 15) / 16;

  // workspace layout (floats): [agg|x1][h|h2][stats][pool]
  float* ws   = (float*)d_ws;
  float* agg  = ws;                              // N*C, later reused as x1
  float* h    = ws + (size_t)N_ * C_DIM;         // N*C, reused as h2
  float* st   = h + (size_t)N_ * C_DIM;
  float* sum1   = st;
  float* sumsq1 = st + G_DIM;
  float* counts = st + 2 * G_DIM;
  float* mean   = st + 3 * G_DIM;
  float* rsig   = st + 4 * G_DIM;
  float* sum2   = st + 5 * G_DIM;
  float* sumsq2 = st + 6 * G_DIM;
  float* pool   = st + 7 * G_DIM;                // G*C
  const int nStats = 7 * G_DIM + G_DIM * C_DIM;

  const int B = 256;

  // 1) zero accumulators
  k_fill0<<<(nStats + B - 1) / B, B, 0, stream>>>(st, nStats);
  // 2) agg = 2*x
  k_init2x<<<(n4 + B - 1) / B, B, 0, stream>>>((const float4*)x, (float4*)agg, n4);
  // 3) agg[dst] += x[src]
  k_scatter<<<((E_ * 16) + B - 1) / B, B, 0, stream>>>((const float4*)x, ei, ei + E_, agg, E_);
  // 4) h = agg @ W_gin + b_gin   (WMMA)
  k_gemm64<<<(nTiles + 7) / 8, B, 0, stream>>>(agg, Wg, bg, h, nTiles);
  // 5) LN1 stats (+counts)
  k_ln_stats<<<(N_ + B - 1) / B, B, 0, stream>>>((const float4*)h, batch, sum1, sumsq1, counts, N_);
  k_ln_finalize<<<1, G_DIM, 0, stream>>>(sum1, sumsq1, counts, mean, rsig);
  // 6) x1 = x + relu(LN1(h))  -> reuse agg buffer
  k_apply_ln1<<<(n4 + B - 1) / B, B, 0, stream>>>((const float4*)x, (const float4*)h, batch,
                                                  mean, rsig, l1w, l1b, (float4*)agg, n4);
  // 7) h2 = x1 @ W_ffn + b_ffn  (WMMA)
  k_gemm64<<<(nTiles + 7) / 8, B, 0, stream>>>(agg, Wf, bf, h, nTiles);
  // 8) LN2 stats
  k_ln_stats<<<(N_ + B - 1) / B, B, 0, stream>>>((const float4*)h, batch, sum2, sumsq2, nullptr, N_);
  k_ln_finalize<<<1, G_DIM, 0, stream>>>(sum2, sumsq2, counts, mean, rsig);
  // 9) x_out = relu(LN2(h2)) -> d_out; pool += x_out
  k_apply_ln2_pool<<<(n4 + B - 1) / B, B, 0, stream>>>((const float4*)h, batch, mean, rsig,
                                                       l2w, l2b, (float4*)out, pool, n4);
  // 10) graph_emds = pool / counts -> tail of d_out
  k_pool_div<<<(G_DIM * C_DIM + B - 1) / B, B, 0, stream>>>(pool, counts,
                                                            out + (size_t)N_ * C_DIM,
                                                            G_DIM * C_DIM);
}